// LinearMap_50285477101809
// MI455X (gfx1250) — compile-verified
//
#include <hip/hip_runtime.h>

#define DIM 512
#define HLEN (DIM * (DIM - 1) / 2)   // 130816 h elements per batch row

typedef __attribute__((ext_vector_type(2))) float v2f;
typedef __attribute__((ext_vector_type(8))) float v8f;

// One block per batch element. 512 threads = 16 waves (wave32).
// Wave w processes 16-row tiles w and (31-w)  (work ~ tile index -> balanced).
// Per tile: D(16x16) accumulated via V_WMMA_F32_16X16X4_F32 with the z-chunk
// broadcast across all 16 B-columns, so every D column equals the row-dot.
// Loop bounds are forced to SGPRs (readfirstlane) so the K-loop is a scalar
// branch: no EXEC-mask churn, EXEC provably all-ones at every WMMA (ISA req).
__global__ void __launch_bounds__(512)
linmap_tri_matvec(const float* __restrict__ z,
                  const float* __restrict__ h,
                  float* __restrict__ out,
                  int batch) {
    __shared__ float zs[DIM];

    const int b   = blockIdx.x;
    const int tid = threadIdx.x;

    // Stage z[b] (2 KB) into LDS once.
    zs[tid] = z[(size_t)b * DIM + tid];
    __syncthreads();

    const int lane = tid & 31;
    const int wave = tid >> 5;            // 0..15
    const int m    = lane & 15;           // row within tile
    const int kb   = (lane >> 4) << 1;    // K pair base: lanes 0-15 -> {0,1}, 16-31 -> {2,3}

    const float* __restrict__ hb = h + (size_t)b * (size_t)HLEN;

    for (int pass = 0; pass < 2; ++pass) {
        const int tile = (pass == 0) ? wave : (31 - wave);
        const int i0   = tile << 4;
        // wave-uniform loop bound -> SGPR, scalar loop control
        const int i0s  = __builtin_amdgcn_readfirstlane(i0);

        const int r      = i0 + m;                 // this lane's matrix row
        const int rowoff = (r * (r - 1)) >> 1;     // tril row base (fits int)
        const float* __restrict__ hrow = hb + rowoff;

        v8f c = {0.f, 0.f, 0.f, 0.f, 0.f, 0.f, 0.f, 0.f};

        // ---- bulk: j in [0, i0): strictly lower for every row in tile.
        // i0 is a multiple of 16 -> process 4 K-chunks (16 cols) per trip,
        // all 8 loads issued before the serial WMMA chain (4 loads in flight).
        for (int j0 = 0; j0 < i0s; j0 += 16) {
            v2f a0, a1, a2, a3, b0, b1, b2, b3;
            const float* __restrict__ hp = hrow + j0 + kb;
            a0.x = hp[0];  a0.y = hp[1];
            a1.x = hp[4];  a1.y = hp[5];
            a2.x = hp[8];  a2.y = hp[9];
            a3.x = hp[12]; a3.y = hp[13];
            const float* __restrict__ zp = zs + j0 + kb;
            b0.x = zp[0];  b0.y = zp[1];
            b1.x = zp[4];  b1.y = zp[5];
            b2.x = zp[8];  b2.y = zp[9];
            b3.x = zp[12]; b3.y = zp[13];
            c = __builtin_amdgcn_wmma_f32_16x16x4_f32(false, a0, false, b0,
                                                      (short)0, c, false, false);
            c = __builtin_amdgcn_wmma_f32_16x16x4_f32(false, a1, false, b1,
                                                      (short)0, c, false, false);
            c = __builtin_amdgcn_wmma_f32_16x16x4_f32(false, a2, false, b2,
                                                      (short)0, c, false, false);
            c = __builtin_amdgcn_wmma_f32_16x16x4_f32(false, a3, false, b3,
                                                      (short)0, c, false, false);
        }

        // ---- diagonal 16x16 block: j in [i0, i0+16), unit diag / zero upper.
        // Branchless: unconditional clamped loads + arithmetic masks, so EXEC
        // stays all-ones (no saveexec around the WMMAs).
        #pragma unroll
        for (int t = 0; t < 4; ++t) {
            const int c0 = i0 + 4 * t + kb;
            const int c1 = c0 + 1;
            int i0c = min(c0, r - 1); i0c = max(i0c, 0);
            int i1c = min(c1, r - 1); i1c = max(i1c, 0);
            const float h0 = hrow[i0c];
            const float h1 = hrow[i1c];
            v2f a, bv;
            a.x = h0 * (float)(c0 < r) + (float)(c0 == r);
            a.y = h1 * (float)(c1 < r) + (float)(c1 == r);
            bv.x = zs[c0];
            bv.y = zs[c1];
            c = __builtin_amdgcn_wmma_f32_16x16x4_f32(false, a, false, bv,
                                                      (short)0, c, false, false);
        }

        // All D columns are equal. Lane 0 holds rows i0..i0+7 in c[0..7],
        // lane 16 holds rows i0+8..i0+15 (16x16 f32 C/D VGPR layout).
        if (lane == 0 || lane == 16) {
            const int rb = i0 + ((lane >> 4) << 3);
            float* __restrict__ o = out + (size_t)b * DIM + rb;
            o[0] = c[0]; o[1] = c[1]; o[2] = c[2]; o[3] = c[3];
            o[4] = c[4]; o[5] = c[5]; o[6] = c[6]; o[7] = c[7];
        }
    }

    // logdet (single zero) appended after f_z.
    if (b == 0 && tid == 0) out[(size_t)batch * DIM] = 0.0f;
}

extern "C" void kernel_launch(void* const* d_in, const int* in_sizes, int n_in,
                              void* d_out, int out_size, void* d_ws, size_t ws_size,
                              hipStream_t stream) {
    const float* z = (const float*)d_in[0];   // (batch, 512) f32
    const float* h = (const float*)d_in[1];   // (batch, 130816) f32
    float* out = (float*)d_out;               // (batch*512 f_z) + 1 logdet

    const int batch = in_sizes[0] / DIM;

    linmap_tri_matvec<<<dim3(batch), dim3(512), 0, stream>>>(z, h, out, batch);
}